// Masked_Model_51264729645285
// MI455X (gfx1250) — compile-verified
//
#include <hip/hip_runtime.h>
#include <stdint.h>

#define IMG   224
#define NPIX  (IMG*IMG)        // 50176
#define NG    (NPIX*3)         // 150528
#define BATCH 256
#define KSEL  12544            // 0.25 * IMG * IMG
#define TILE  12544            // NG / 12 exactly
#define NTILES 12
#define EQCAP 1024
#define MASKW (NPIX/32)        // 1568 words per sample

typedef unsigned int uint32;
typedef uint32 v4u __attribute__((ext_vector_type(4)));
typedef int    v8i __attribute__((ext_vector_type(8)));
typedef int    v4i __attribute__((ext_vector_type(4)));

// Monotonic total-order key: larger float -> larger uint key.
__device__ __forceinline__ uint32 fkey(float f) {
  uint32 u = __float_as_uint(f);
  return (u & 0x80000000u) ? ~u : (u | 0x80000000u);
}

// ---------------------------------------------------------------------------
// Pass 1: unfiltered histogram of key[31:21] (2048 bins).
// 256-thread block (8 waves). Wave 0 issues a TDM tensor_load_to_lds DMA of
// the 12544-float tile via a scalar-uniform branch (tensor ops ignore EXEC,
// so a divergent branch would duplicate the DMA 8x), waits TENSORcnt==0,
// then the workgroup barrier releases all 8 waves to histogram from LDS.
// ---------------------------------------------------------------------------
__global__ __launch_bounds__(256)
void hist_l1_tdm(const float* __restrict__ grad, uint32* __restrict__ ghist) {
  __shared__ float  tile[TILE];     // 49 KB
  __shared__ uint32 lhist[2048];    //  8 KB
  const int b   = blockIdx.y;
  const int t   = blockIdx.x;
  const int tid = threadIdx.x;

  for (int i = tid; i < 2048; i += 256) lhist[i] = 0;

#if __has_builtin(__builtin_amdgcn_tensor_load_to_lds)
  // Wave-uniform scalar condition -> s_cbranch, only wave 0 issues the DMA.
  int wtid = __builtin_amdgcn_readfirstlane((int)threadIdx.x);
  if (wtid == 0) {
    unsigned long long gaddr =
        (unsigned long long)(uintptr_t)(grad + (size_t)b * NG + (size_t)t * TILE);
    uint32 laddr = (uint32)(unsigned long long)(uintptr_t)&tile[0];

    v4u g0;
    g0[0] = 1u;                                   // count=1, user mode
    g0[1] = laddr;                                // lds_addr
    g0[2] = (uint32)gaddr;                        // global_addr[31:0]
    g0[3] = (uint32)((gaddr >> 32) & 0x1FFFFFFu)  // global_addr[56:32]
            | (2u << 30);                         // type=2 (image)

    v8i g1;
    g1[0] = (int)(2u << 16);                        // data_size=4B; no iterate/pad
    g1[1] = (int)(((uint32)TILE & 0xFFFFu) << 16);  // tensor_dim0[15:0]
    g1[2] = (int)(((uint32)TILE >> 16) & 0xFFFFu);  // tensor_dim0[31:16]
    g1[3] = (int)(((uint32)TILE & 0xFFFFu) << 16);  // tile_dim0 = TILE
    g1[4] = 0;                                      // tile_dim1 = tile_dim2 = 0 (1D)
    g1[5] = (int)TILE;                              // tensor_dim0_stride lo32
    g1[6] = 0;
    g1[7] = 0;

    v4i z4; z4[0] = z4[1] = z4[2] = z4[3] = 0;
    v8i z8; z8[0] = z8[1] = z8[2] = z8[3] = z8[4] = z8[5] = z8[6] = z8[7] = 0;
    __builtin_amdgcn_tensor_load_to_lds(g0, g1, z4, z4, z8, 0);
    __builtin_amdgcn_s_wait_tensorcnt(0);
  }
#else
  for (int i = tid; i < TILE; i += 256)
    tile[i] = grad[(size_t)b * NG + (size_t)t * TILE + i];
#endif
  __syncthreads();

  for (int i = tid; i < TILE; i += 256) {
    uint32 k = fkey(tile[i]);
    atomicAdd(&lhist[k >> 21], 1u);
  }
  __syncthreads();

  uint32* gh = ghist + (size_t)b * 2048;
  for (int i = tid; i < 2048; i += 256) {
    uint32 v = lhist[i];
    if (v) atomicAdd(&gh[i], v);
  }
}

// ---------------------------------------------------------------------------
// Passes 2/3: histogram of next key bits, filtered by current prefix.
// float4-vectorized streaming reads (NG % 4 == 0).
// ---------------------------------------------------------------------------
__global__ __launch_bounds__(256)
void hist_filtered(const float* __restrict__ grad, uint32* __restrict__ ghist,
                   const uint32* __restrict__ prefix, int nbins, int shift,
                   int topShift) {
  __shared__ uint32 lhist[2048];
  const int b   = blockIdx.y;
  const int tid = threadIdx.x;
  for (int i = tid; i < nbins; i += 256) lhist[i] = 0;
  __syncthreads();

  const uint32 pfx  = prefix[b];
  const uint32 bmsk = (uint32)(nbins - 1);
  const float* g = grad + (size_t)b * NG;
  const int NQ = NG / 4;
  for (int q = blockIdx.x * 256 + tid; q < NQ; q += gridDim.x * 256) {
    float4 v = *(const float4*)(g + (size_t)q * 4);
    uint32 k0 = fkey(v.x), k1 = fkey(v.y), k2 = fkey(v.z), k3 = fkey(v.w);
    if ((k0 >> topShift) == pfx) atomicAdd(&lhist[(k0 >> shift) & bmsk], 1u);
    if ((k1 >> topShift) == pfx) atomicAdd(&lhist[(k1 >> shift) & bmsk], 1u);
    if ((k2 >> topShift) == pfx) atomicAdd(&lhist[(k2 >> shift) & bmsk], 1u);
    if ((k3 >> topShift) == pfx) atomicAdd(&lhist[(k3 >> shift) & bmsk], 1u);
  }
  __syncthreads();

  uint32* gh = ghist + (size_t)b * 2048;
  for (int i = tid; i < nbins; i += 256) {
    uint32 v = lhist[i];
    if (v) atomicAdd(&gh[i], v);
  }
}

// ---------------------------------------------------------------------------
// Per-sample scan from high bin to low: find bin containing the Krem-th
// largest; update prefix = (prefix<<bits)|bin and Krem -= count(above bin).
// ---------------------------------------------------------------------------
__global__ __launch_bounds__(256)
void scan_level(const uint32* __restrict__ ghist, uint32* __restrict__ prefix,
                uint32* __restrict__ krem, int nbins, int bits) {
  __shared__ uint32 csum[256];
  const int b   = blockIdx.x;
  const int tid = threadIdx.x;
  const uint32* h = ghist + (size_t)b * 2048;
  const int per = nbins / 256;

  uint32 s = 0;
  for (int j = 0; j < per; ++j) s += h[tid * per + j];
  csum[tid] = s;
  __syncthreads();

  if (tid == 0) {
    uint32 Krem = krem[b];
    uint32 cum  = 0;
    int chunk = 0;
    for (int c = 255; c >= 0; --c) {
      if (cum + csum[c] >= Krem) { chunk = c; break; }
      cum += csum[c];
    }
    int bin = chunk * per;
    for (int i = per - 1; i >= 0; --i) {
      uint32 v = h[chunk * per + i];
      if (cum + v >= Krem) { bin = chunk * per + i; break; }
      cum += v;
    }
    prefix[b] = (prefix[b] << bits) | (uint32)bin;
    krem[b]   = Krem - cum;   // #equal-valued elements to take (>=1)
  }
}

// ---------------------------------------------------------------------------
// Merged pass: set mask bits for key > T, and collect tie candidates
// (key == T) for rank-based resolution. One full gradient read saved vs
// doing these in two separate passes. float4-vectorized.
// ---------------------------------------------------------------------------
__global__ __launch_bounds__(256)
void mask_and_collect(const float* __restrict__ grad,
                      const uint32* __restrict__ prefix,
                      uint32* __restrict__ eqCnt, uint32* __restrict__ eqList,
                      uint32* __restrict__ mask) {
  const int b = blockIdx.y;
  const uint32 T = prefix[b];
  const float* g = grad + (size_t)b * NG;
  uint32* m = mask + (size_t)b * MASKW;
  const int NQ = NG / 4;
  for (int q = blockIdx.x * 256 + threadIdx.x; q < NQ; q += gridDim.x * 256) {
    float4 v = *(const float4*)(g + (size_t)q * 4);
    uint32 kk[4] = { fkey(v.x), fkey(v.y), fkey(v.z), fkey(v.w) };
    const int i0 = q * 4;
#pragma unroll
    for (int e = 0; e < 4; ++e) {
      uint32 k = kk[e];
      int i = i0 + e;
      if (k > T) {
        int p = i % NPIX;               // reference: idx mod (IMG*IMG)
        atomicOr(&m[p >> 5], 1u << (p & 31));
      } else if (k == T) {
        uint32 pos = atomicAdd(&eqCnt[b], 1u);
        if (pos < EQCAP) eqList[(size_t)b * EQCAP + pos] = (uint32)i;
      }
    }
  }
}

// ---------------------------------------------------------------------------
// Resolve ties: among indices with key == T, select the `need` smallest
// (matches jax.lax.top_k tie-breaking) via order-independent rank counting,
// and set their mask bits directly. Deterministic.
// ---------------------------------------------------------------------------
__global__ __launch_bounds__(256)
void resolve_ties(const uint32* __restrict__ eqCnt, const uint32* __restrict__ eqList,
                  const uint32* __restrict__ krem, uint32* __restrict__ mask) {
  __shared__ uint32 sl[EQCAP];
  const int b = blockIdx.x;
  uint32 cnt = eqCnt[b];
  int c = (cnt < (uint32)EQCAP) ? (int)cnt : EQCAP;
  const uint32 need = krem[b];
  for (int i = threadIdx.x; i < c; i += 256) sl[i] = eqList[(size_t)b * EQCAP + i];
  __syncthreads();
  uint32* m = mask + (size_t)b * MASKW;
  for (int j = threadIdx.x; j < c; j += 256) {
    uint32 v = sl[j];
    uint32 rank = 0;
    for (int t = 0; t < c; ++t) rank += (sl[t] < v) ? 1u : 0u;
    if (rank < need) {
      int p = (int)(v % (uint32)NPIX);
      atomicOr(&m[p >> 5], 1u << (p & 31));
    }
  }
}

// ---------------------------------------------------------------------------
// Coalesced masked copy: 4 pixels (12 floats = 3x float4) per thread.
// ---------------------------------------------------------------------------
__global__ __launch_bounds__(256)
void write_out(const float* __restrict__ data, const uint32* __restrict__ mask,
               float* __restrict__ out) {
  const int t = blockIdx.x * 256 + threadIdx.x;       // quad id
  const int quadsPerSample = NPIX / 4;                // 12544
  const int b = t / quadsPerSample;
  const int q = t - b * quadsPerSample;
  if (b >= BATCH) return;
  const int p0 = q * 4;
  const uint32 bits = (mask[(size_t)b * MASKW + (p0 >> 5)] >> (p0 & 31)) & 0xFu;
  const size_t base = (size_t)b * NG + (size_t)p0 * 3;   // 16B aligned
  float4 v0 = *(const float4*)(data + base);
  float4 v1 = *(const float4*)(data + base + 4);
  float4 v2 = *(const float4*)(data + base + 8);
  if (bits & 1u) { v0.x = 0.f; v0.y = 0.f; v0.z = 0.f; }
  if (bits & 2u) { v0.w = 0.f; v1.x = 0.f; v1.y = 0.f; }
  if (bits & 4u) { v1.z = 0.f; v1.w = 0.f; v2.x = 0.f; }
  if (bits & 8u) { v2.y = 0.f; v2.z = 0.f; v2.w = 0.f; }
  *(float4*)(out + base)     = v0;
  *(float4*)(out + base + 4) = v1;
  *(float4*)(out + base + 8) = v2;
}

__global__ void init_state(uint32* krem, uint32* prefix, uint32* eqCnt) {
  int i = threadIdx.x;
  if (i < BATCH) {
    krem[i]   = KSEL;
    prefix[i] = 0;
    eqCnt[i]  = 0;
  }
}

extern "C" void kernel_launch(void* const* d_in, const int* in_sizes, int n_in,
                              void* d_out, int out_size, void* d_ws, size_t ws_size,
                              hipStream_t stream) {
  const float* data = (const float*)d_in[0];   // [256,224,224,3]
  const float* grad = (const float*)d_in[1];   // [256,150528]
  float* out = (float*)d_out;

  // Workspace layout (bytes)
  const size_t HIST_OFF   = 0;                         // 2,097,152
  const size_t HIST_BYTES = (size_t)BATCH * 2048 * 4;
  const size_t KREM_OFF   = HIST_OFF + HIST_BYTES;
  const size_t PFX_OFF    = KREM_OFF + 1024;
  const size_t EQC_OFF    = PFX_OFF + 1024;
  const size_t EQL_OFF    = EQC_OFF + 1024;            // 1 MB
  const size_t MASK_OFF   = EQL_OFF + (size_t)BATCH * EQCAP * 4;
  const size_t MASK_BYTES = (size_t)BATCH * MASKW * 4; // 1,605,632

  char* ws = (char*)d_ws;
  uint32* hist   = (uint32*)(ws + HIST_OFF);
  uint32* krem   = (uint32*)(ws + KREM_OFF);
  uint32* prefix = (uint32*)(ws + PFX_OFF);
  uint32* eqCnt  = (uint32*)(ws + EQC_OFF);
  uint32* eqList = (uint32*)(ws + EQL_OFF);
  uint32* mask   = (uint32*)(ws + MASK_OFF);
  (void)in_sizes; (void)n_in; (void)out_size; (void)ws_size;

  (void)hipMemsetAsync(hist, 0, HIST_BYTES, stream);
  (void)hipMemsetAsync(mask, 0, MASK_BYTES, stream);
  init_state<<<1, 256, 0, stream>>>(krem, prefix, eqCnt);

  // Level 1: key[31:21], 2048 bins, TDM-staged tiles.
  hist_l1_tdm<<<dim3(NTILES, BATCH), 256, 0, stream>>>(grad, hist);
  scan_level<<<BATCH, 256, 0, stream>>>(hist, prefix, krem, 2048, 11);

  // Level 2: key[20:10], 2048 bins, filter key>>21 == prefix.
  (void)hipMemsetAsync(hist, 0, HIST_BYTES, stream);
  hist_filtered<<<dim3(16, BATCH), 256, 0, stream>>>(grad, hist, prefix, 2048, 10, 21);
  scan_level<<<BATCH, 256, 0, stream>>>(hist, prefix, krem, 2048, 11);

  // Level 3: key[9:0], 1024 bins, filter key>>10 == prefix.
  (void)hipMemsetAsync(hist, 0, HIST_BYTES, stream);
  hist_filtered<<<dim3(16, BATCH), 256, 0, stream>>>(grad, hist, prefix, 1024, 0, 10);
  scan_level<<<BATCH, 256, 0, stream>>>(hist, prefix, krem, 1024, 10);
  // prefix[b] = exact 32-bit threshold key; krem[b] = #equal-valued to take.

  mask_and_collect<<<dim3(16, BATCH), 256, 0, stream>>>(grad, prefix, eqCnt, eqList, mask);
  resolve_ties<<<BATCH, 256, 0, stream>>>(eqCnt, eqList, krem, mask);
  write_out<<<(BATCH * (NPIX / 4) + 255) / 256, 256, 0, stream>>>(data, mask, out);
}